// PALM_68298569941465
// MI455X (gfx1250) — compile-verified
//
#include <hip/hip_runtime.h>
#include <hip/hip_bf16.h>

// ---------------------------------------------------------------------------
// PALM loss on gfx1250 (MI455X).
//  features [65536x128] f32, protos [500x128] f32, targets [65536] i32 -> scalar f32
//  Heavy GEMMs use V_WMMA_F32_16X16X4_F32 (exact f32 path; exp(S/0.05) is too
//  error-amplifying for bf16 WMMA). Score matrix S [65536x512-padded] lives in
//  workspace (134 MB, fits MI455X's 192 MB L2); Sinkhorn reduced to scaling
//  vectors u[500], v[65536]. GEMM blocked 16x64 per wave: one A fragment feeds
//  4 WMMAs (1.25 loads/WMMA instead of 2).
// ---------------------------------------------------------------------------

#define NN   65536
#define DD   128
#define PREAL 500
#define PP   512          // padded proto count (tiles of 16)
#define NCLS 10
#define KTOP 5
#define INV_EPS   20.0f   // 1/0.05
#define INV_TEMP  10.0f   // 1/0.1

typedef float v2f __attribute__((ext_vector_type(2)));
typedef float v8f __attribute__((ext_vector_type(8)));

// ---------------- workspace layout (floats) --------------------------------
static const size_t oS  = 0;                          // NN*PP scores
static const size_t oU  = oS  + (size_t)NN * PP;      // PP   proto scaling u
static const size_t oV  = oU  + PP;                   // NN   sample scaling v
static const size_t oRP = oV  + NN;                   // 256*PP row partials
static const size_t oTW = oRP + 256 * PP;             // NN*5 topk weights
static const size_t oTI = oTW + (size_t)NN * KTOP;    // NN*5 topk indices (int)
static const size_t oCW = oTI + (size_t)NN * KTOP;    // PP   column sums
static const size_t oUF = oCW + PP;                   // PP*DD update_features
static const size_t oPN = oUF + (size_t)PP * DD;      // PP*DD protos_new
static const size_t oPC = oPN + (size_t)PP * DD;      // PP   per-proto pc rows
static const size_t oPT = oPC + PP;                   // 8192 pos-neg partials

// ---------------- utility kernels ------------------------------------------
__global__ void palm_fill(float* __restrict__ p, float val, int n) {
  int i = blockIdx.x * blockDim.x + threadIdx.x;
  if (i < n) p[i] = val;
}

// ---------------- GEMM: S[n,p] = sum_d A[n,d]*B[p,d] -----------------------
// One wave computes a 16x64 strip: 4 proto tiles share each A fragment.
// 128 WMMAs per wave (K=128 in steps of 4, 4 tiles wide).
__global__ void palm_gemm(const float* __restrict__ A, const float* __restrict__ B,
                          float* __restrict__ S) {
  const int lane = threadIdx.x & 31;
  const int wave = threadIdx.x >> 5;
  const int tile = blockIdx.x * 8 + wave;      // (NN/16)*(PP/64) strips total
  const int tpq  = PP / 64;                    // 8 strips across protos
  const int m0   = (tile / tpq) * 16;
  const int p0   = (tile % tpq) * 64;
  const int half = lane >> 4;                  // 0/1 selects K pair {0,1}/{2,3}
  const int lr   = lane & 15;

  const float* ap = A + (size_t)(m0 + lr) * DD + half * 2;
  const float* bp[4];
  float bm[4];
  #pragma unroll
  for (int t = 0; t < 4; ++t) {
    int brow = p0 + t * 16 + lr;
    bool bv = (brow < PREAL);
    bm[t] = bv ? 1.0f : 0.0f;                  // zero padded proto rows
    bp[t] = B + (size_t)(bv ? brow : 0) * DD + half * 2;
  }

  v8f c0 = {0.f,0.f,0.f,0.f,0.f,0.f,0.f,0.f};
  v8f c1 = c0, c2 = c0, c3 = c0;
  #pragma unroll 2
  for (int kk = 0; kk < DD; kk += 4) {
    v2f a, b0, b1, b2, b3;
    a.x  = ap[kk];              a.y  = ap[kk + 1];
    b0.x = bp[0][kk] * bm[0];   b0.y = bp[0][kk + 1] * bm[0];
    b1.x = bp[1][kk] * bm[1];   b1.y = bp[1][kk + 1] * bm[1];
    b2.x = bp[2][kk] * bm[2];   b2.y = bp[2][kk + 1] * bm[2];
    b3.x = bp[3][kk] * bm[3];   b3.y = bp[3][kk + 1] * bm[3];
    c0 = __builtin_amdgcn_wmma_f32_16x16x4_f32(false, a, false, b0, (short)0, c0, false, false);
    c1 = __builtin_amdgcn_wmma_f32_16x16x4_f32(false, a, false, b1, (short)0, c1, false, false);
    c2 = __builtin_amdgcn_wmma_f32_16x16x4_f32(false, a, false, b2, (short)0, c2, false, false);
    c3 = __builtin_amdgcn_wmma_f32_16x16x4_f32(false, a, false, b3, (short)0, c3, false, false);
  }

  // D layout: VGPR r -> M = r + 8*half, N = lane&15
  auto store_tile = [&](const v8f& c, int t) {
    int pcol = p0 + t * 16 + lr;
    bool pv = (pcol < PREAL);
    #pragma unroll
    for (int r = 0; r < 8; ++r) {
      int m = m0 + r + half * 8;
      S[(size_t)m * PP + pcol] = pv ? c[r] : -1e30f;  // pad -> exp()==0
    }
  };
  store_tile(c0, 0); store_tile(c1, 1); store_tile(c2, 2); store_tile(c3, 3);
}

// ---------------- Sinkhorn: row partials, u, v ------------------------------
// rowpart[chunk][p] = sum_{n in chunk} exp(S[n,p]/eps) * v[n]
__global__ void palm_rowpart(const float* __restrict__ S, const float* __restrict__ v,
                             float* __restrict__ rp) {
  int p = threadIdx.x;            // 512
  int n0 = blockIdx.x * 256;      // 256 chunks
  float sum = 0.f;
  for (int i = 0; i < 256; ++i) {
    float s = S[(size_t)(n0 + i) * PP + p];
    sum += __expf(s * INV_EPS) * v[n0 + i];
  }
  rp[blockIdx.x * PP + p] = sum;
}

// u[p] = (1/500) / max(sum_c rp[c][p], 1e-12)   (fixed-order, deterministic)
__global__ void palm_urow(const float* __restrict__ rp, float* __restrict__ u) {
  int p = threadIdx.x;            // 512, single block
  float rs = 0.f;
  for (int cchunk = 0; cchunk < 256; ++cchunk) rs += rp[cchunk * PP + p];
  u[p] = (p < PREAL) ? (1.0f / (float)PREAL) / fmaxf(rs, 1e-12f) : 0.0f;
}

// v[n] = (1/N) / max(sum_p exp(S[n,p]/eps)*u[p], 1e-12)  (wave per n)
__global__ void palm_colv(const float* __restrict__ S, const float* __restrict__ u,
                          float* __restrict__ v) {
  int lane = threadIdx.x & 31;
  int n = blockIdx.x * 8 + (threadIdx.x >> 5);
  const float* row = S + (size_t)n * PP;
  float sum = 0.f;
  for (int p = lane; p < PP; p += 32) sum += __expf(row[p] * INV_EPS) * u[p];
  for (int o = 16; o; o >>= 1) sum += __shfl_xor(sum, o, 32);
  if (lane == 0) v[n] = (1.0f / (float)NN) / fmaxf(sum, 1e-12f);
}

// ---------------- top-5 over the 50 same-class protos (wave per n) ----------
__global__ void palm_topk(const float* __restrict__ S, const float* __restrict__ u,
                          const int* __restrict__ targets, int* __restrict__ ti,
                          float* __restrict__ tw, float* __restrict__ colw) {
  int lane = threadIdx.x & 31;
  int n = blockIdx.x * 8 + (threadIdx.x >> 5);
  int t = targets[n];
  const float* row = S + (size_t)n * PP;
  // candidates j=0..49 -> proto p = t + 10*j ; value = exp(S/eps)*u[p]
  // (the common positive factor N*v[n] does not change ranking or row-l1)
  int j0 = lane, j1 = lane + 32;
  float x0 = -1.f, x1 = -1.f;
  if (j0 < 50) { int p = t + NCLS * j0; x0 = __expf(row[p] * INV_EPS) * u[p]; }
  if (j1 < 50) { int p = t + NCLS * j1; x1 = __expf(row[p] * INV_EPS) * u[p]; }
  float wv[KTOP]; int wj[KTOP];
  #pragma unroll
  for (int k = 0; k < KTOP; ++k) {
    float bv = x0; int bj = j0;
    if (x1 > bv) { bv = x1; bj = j1; }
    for (int o = 16; o; o >>= 1) {
      float ov = __shfl_xor(bv, o, 32);
      int   oj = __shfl_xor(bj, o, 32);
      if (ov > bv || (ov == bv && oj < bj)) { bv = ov; bj = oj; }
    }
    wv[k] = bv; wj[k] = bj;
    if (bj == j0) x0 = -1.f;
    if (bj == j1) x1 = -1.f;
  }
  float rs = wv[0] + wv[1] + wv[2] + wv[3] + wv[4];
  float inv = 1.0f / fmaxf(rs, 1e-12f);
  #pragma unroll
  for (int k = 0; k < KTOP; ++k) {
    if (lane == k) {
      int p = t + NCLS * wj[k];
      float wgt = wv[k] * inv;
      ti[(size_t)n * KTOP + k] = p;
      tw[(size_t)n * KTOP + k] = wgt;
      atomicAdd(&colw[p], wgt);
    }
  }
}

// update_features[p,d] += (w / colsum[p]) * features[n,d]
__global__ void palm_uf(const float* __restrict__ feats, const int* __restrict__ ti,
                        const float* __restrict__ tw, const float* __restrict__ colw,
                        float* __restrict__ UF) {
  int n = blockIdx.x;       // 65536
  int d = threadIdx.x;      // 128
  float f = feats[(size_t)n * DD + d];
  #pragma unroll
  for (int k = 0; k < KTOP; ++k) {
    int p = ti[(size_t)n * KTOP + k];
    float wc = tw[(size_t)n * KTOP + k] / fmaxf(colw[p], 1e-12f);
    atomicAdd(&UF[(size_t)p * DD + d], wc * f);
  }
}

// protos_new = l2rows(0.99*protos + 0.01*UF)
__global__ void palm_pnew(const float* __restrict__ protos, const float* __restrict__ UF,
                          float* __restrict__ pn) {
  int p = blockIdx.x;       // 500
  int d = threadIdx.x;      // 128
  float val = 0.99f * protos[(size_t)p * DD + d] + 0.01f * UF[(size_t)p * DD + d];
  float ss = val * val;
  for (int o = 16; o; o >>= 1) ss += __shfl_xor(ss, o, 32);
  __shared__ float ls[4];
  if ((threadIdx.x & 31) == 0) ls[threadIdx.x >> 5] = ss;
  __syncthreads();
  float tot = ls[0] + ls[1] + ls[2] + ls[3];
  pn[(size_t)p * DD + d] = val / fmaxf(sqrtf(tot), 1e-12f);
}

// pos[n]-neg[n] per sample (wave per n), block partial sums -> partial[]
__global__ void palm_posneg(const float* __restrict__ S, const float* __restrict__ u,
                            const int* __restrict__ ti, float* __restrict__ partial) {
  __shared__ float lw[8];
  int lane = threadIdx.x & 31;
  int wave = threadIdx.x >> 5;
  int n = blockIdx.x * 8 + wave;
  const float* row = S + (size_t)n * PP;
  // neg = log(sum_p exp(S/temp)); padded cols are -1e30 -> exp = 0
  float ns = 0.f;
  for (int p = lane; p < PP; p += 32) ns += __expf(row[p] * INV_TEMP);
  for (int o = 16; o; o >>= 1) ns += __shfl_xor(ns, o, 32);
  // pos: row-l1 of (exp(S/eps)*u2) at the stored 5 indices, dot with S/temp
  float q = 0.f, s5 = 0.f;
  if (lane < KTOP) {
    int p = ti[(size_t)n * KTOP + lane];
    s5 = row[p];
    q = __expf(s5 * INV_EPS) * u[p];
  }
  float qs = q;
  for (int o = 16; o; o >>= 1) qs += __shfl_xor(qs, o, 32);
  float posl = (lane < KTOP) ? (q / fmaxf(qs, 1e-12f)) * (s5 * INV_TEMP) : 0.f;
  for (int o = 16; o; o >>= 1) posl += __shfl_xor(posl, o, 32);
  if (lane == 0) lw[wave] = posl - __logf(ns);
  __syncthreads();
  if (threadIdx.x == 0) {
    float s = 0.f;
    for (int w = 0; w < 8; ++w) s += lw[w];
    partial[blockIdx.x] = s;
  }
}

// proto-contrast per row i: pos2 - neg2 (block per i, 512 threads)
__global__ void palm_pc(const float* __restrict__ pn, float* __restrict__ pcrow) {
  __shared__ float pi[DD];
  __shared__ float red[512];
  int i = blockIdx.x;       // 500
  int j = threadIdx.x;      // 512
  if (j < DD) pi[j] = pn[(size_t)i * DD + j];
  __syncthreads();
  float a = -3.0e38f;
  if (j < PREAL) {
    float dot = 0.f;
    for (int d = 0; d < DD; ++d) dot += pi[d] * pn[(size_t)j * DD + d];
    a = dot * 2.0f;                                  // /0.5
  }
  red[j] = a; __syncthreads();
  for (int s = 256; s; s >>= 1) { if (j < s) red[j] = fmaxf(red[j], red[j + s]); __syncthreads(); }
  float mx = red[0]; __syncthreads();
  float lg = a - mx;
  bool valid = (j < PREAL) && (j != i);
  bool same  = valid && ((j % NCLS) == (i % NCLS));
  red[j] = same ? lg : 0.f; __syncthreads();
  for (int s = 256; s; s >>= 1) { if (j < s) red[j] += red[j + s]; __syncthreads(); }
  float possum = red[0]; __syncthreads();
  red[j] = valid ? __expf(lg) : 0.f; __syncthreads();
  for (int s = 256; s; s >>= 1) { if (j < s) red[j] += red[j + s]; __syncthreads(); }
  float negsum = red[0];
  if (j == 0) pcrow[i] = possum * (1.0f / 49.0f) - __logf(negsum);
}

// final: out = -(mean(pos-neg)) - mean(pcrow)
__global__ void palm_final(const float* __restrict__ partial, const float* __restrict__ pcrow,
                           float* __restrict__ out) {
  __shared__ float red[256];
  int t = threadIdx.x;
  float s1 = 0.f, s2 = 0.f;
  for (int i = t; i < 8192; i += 256) s1 += partial[i];
  for (int i = t; i < PREAL; i += 256) s2 += pcrow[i];
  red[t] = s1; __syncthreads();
  for (int s = 128; s; s >>= 1) { if (t < s) red[t] += red[t + s]; __syncthreads(); }
  s1 = red[0]; __syncthreads();
  red[t] = s2; __syncthreads();
  for (int s = 128; s; s >>= 1) { if (t < s) red[t] += red[t + s]; __syncthreads(); }
  s2 = red[0];
  if (t == 0) out[0] = -(s1 / (float)NN) - (s2 / (float)PREAL);
}

// ---------------------------------------------------------------------------
extern "C" void kernel_launch(void* const* d_in, const int* in_sizes, int n_in,
                              void* d_out, int out_size, void* d_ws, size_t ws_size,
                              hipStream_t stream) {
  (void)in_sizes; (void)n_in; (void)out_size; (void)ws_size;
  const float* features = (const float*)d_in[0];
  const float* protos   = (const float*)d_in[1];
  const int*   targets  = (const int*)d_in[2];
  float* out = (float*)d_out;
  float* W   = (float*)d_ws;   // needs ~138 MB

  float* S  = W + oS;
  float* U  = W + oU;
  float* V  = W + oV;
  float* RP = W + oRP;
  float* TW = W + oTW;
  int*   TI = (int*)(W + oTI);
  float* CW = W + oCW;
  float* UF = W + oUF;
  float* PN = W + oPN;
  float* PC = W + oPC;
  float* PT = W + oPT;

  const int gemmBlocks = (NN / 16) * (PP / 64) / 8;   // 4096 blocks, 8 waves each

  // ---- S = features @ protos^T ----
  palm_gemm<<<gemmBlocks, 256, 0, stream>>>(features, protos, S);

  // ---- Sinkhorn #1 -> u, v ----
  palm_fill<<<NN / 256, 256, 0, stream>>>(V, 1.0f, NN);
  for (int it = 0; it < 3; ++it) {
    palm_rowpart<<<256, PP, 0, stream>>>(S, V, RP);
    palm_urow<<<1, PP, 0, stream>>>(RP, U);
    palm_colv<<<NN / 8, 256, 0, stream>>>(S, U, V);
  }

  // ---- top-5 mask, row-l1 + col sums, update_features ----
  palm_fill<<<(PP + PP * DD + 255) / 256, 256, 0, stream>>>(CW, 0.0f, PP + PP * DD);
  palm_topk<<<NN / 8, 256, 0, stream>>>(S, U, targets, TI, TW, CW);
  palm_uf<<<NN, DD, 0, stream>>>(features, TI, TW, CW, UF);
  palm_pnew<<<PREAL, DD, 0, stream>>>(protos, UF, PN);

  // ---- S = features @ protos_new^T (overwrite) ----
  palm_gemm<<<gemmBlocks, 256, 0, stream>>>(features, PN, S);

  // ---- Sinkhorn #2 -> u2 (v2 cancels in row-l1) ----
  palm_fill<<<NN / 256, 256, 0, stream>>>(V, 1.0f, NN);
  for (int it = 0; it < 3; ++it) {
    palm_rowpart<<<256, PP, 0, stream>>>(S, V, RP);
    palm_urow<<<1, PP, 0, stream>>>(RP, U);
    palm_colv<<<NN / 8, 256, 0, stream>>>(S, U, V);
  }

  // ---- mle pos/neg + proto-contrast + combine ----
  palm_posneg<<<NN / 8, 256, 0, stream>>>(S, U, TI, PT);
  palm_pc<<<PREAL, PP, 0, stream>>>(PN, PC);
  palm_final<<<1, 256, 0, stream>>>(PT, PC, out);
}